// ROI_pooling_62517543961174
// MI455X (gfx1250) — compile-verified
//
#include <hip/hip_runtime.h>
#include <cstdint>

// Problem constants (match reference setup_inputs / _roi_pool)
#define BB   4
#define CC   256
#define HH   64
#define WW   64
#define RR   256
#define HWN  (HH * WW)           // 4096 positions
#define POOLN 7
#define RESZ  14
#define KMAX  15                 // worst-case footprint per axis (see analysis)
#define LDS_FLOATS (KMAX * KMAX * CC)   // 57600 floats
#define LDS_BYTES  (LDS_FLOATS * 4)     // 230400 B  (< 320 KB WGP LDS)

typedef unsigned int u32x4 __attribute__((ext_vector_type(4)));
typedef int          i32x8 __attribute__((ext_vector_type(8)));
typedef int          i32x4 __attribute__((ext_vector_type(4)));

// ---------------- CDNA5 async global->LDS helpers -------------------------
// LDS address comes from the VDST VGPR plus offset; flat LDS pointers carry the
// LDS byte offset in their low 32 bits (ISA 10.2 aperture table), so truncate.
__device__ __forceinline__ unsigned lds_off_u32(const void* p) {
  return (unsigned)(uintptr_t)p;
}
__device__ __forceinline__ void async_ld_b128(unsigned lds_off, unsigned long long gaddr) {
  asm volatile("global_load_async_to_lds_b128 %0, %1, off"
               :: "v"(lds_off), "v"(gaddr) : "memory");
}
__device__ __forceinline__ void async_ld_b32(unsigned lds_off, unsigned long long gaddr) {
  asm volatile("global_load_async_to_lds_b32 %0, %1, off"
               :: "v"(lds_off), "v"(gaddr) : "memory");
}
__device__ __forceinline__ void wait_asynccnt0() {
#if __has_builtin(__builtin_amdgcn_s_wait_asynccnt)
  __builtin_amdgcn_s_wait_asynccnt(0);
#else
  asm volatile("s_wait_asynccnt 0" ::: "memory");
#endif
}
__device__ __forceinline__ void wait_tensorcnt0() {
#if __has_builtin(__builtin_amdgcn_s_wait_tensorcnt)
  __builtin_amdgcn_s_wait_tensorcnt(0);
#else
  asm volatile("s_wait_tensorcnt 0" ::: "memory");
#endif
}

#if __has_builtin(__builtin_amdgcn_tensor_load_to_lds)
#define HAVE_TDM 1
#else
#define HAVE_TDM 0
#endif

// ---------------- BCHW -> BHWC transpose (coalesced both sides) -----------
__global__ __launch_bounds__(256)
void transpose_bchw_to_bhwc(const float* __restrict__ src, float* __restrict__ dst) {
  __shared__ float tile[16][17];           // +1 pad: bank-conflict free
  const int b  = blockIdx.z;
  const int c0 = blockIdx.y << 4;          // channel tile
  const int p0 = blockIdx.x << 4;          // HW-position tile
  const int tx = threadIdx.x, ty = threadIdx.y;
  // read:  src[b][c0+ty][p0+tx]   (coalesced along HW)
  tile[ty][tx] = src[(((size_t)b * CC + (c0 + ty)) << 12) + (p0 + tx)];
  __syncthreads();
  // write: dst[b][p0+ty][c0+tx]   (coalesced along C)
  dst[((((size_t)b << 12) + (p0 + ty)) << 8) + (c0 + tx)] = tile[tx][ty];
}

// ---------------- ROI-align + maxpool, one workgroup per ROI --------------
// CHLAST=true  : src is BHWC (transposed) -> one TDM tensor_load_to_lds
//                (2D tile: Ky rows x (Kx*C) contiguous floats, row stride W*C)
// CHLAST=false : src is BCHW              -> per-lane b32 async gathers
template <bool CHLAST>
__global__ __launch_bounds__(256)
void roi_align_kernel(const float* __restrict__ src,
                      const float* __restrict__ rois,
                      float* __restrict__ out) {
  extern __shared__ float smem[];          // [Ky][Kx][C] staged footprint
  const int tid = threadIdx.x;             // == channel
  const int br  = blockIdx.x;              // b*R + r
  const int b   = br >> 8;                 // R == 256

  // ---- ROI math (mirrors reference exactly; truncation toward zero) ----
  const float4 roi = ((const float4*)rois)[br];   // (cx, cy, w, h) in image px
  const int cx = (int)(roi.x * 0.0625f);
  const int cy = (int)(roi.y * 0.0625f);
  const int w2 = max(((int)(roi.z * 0.0625f)) >> 1, 1);
  const int h2 = max(((int)(roi.w * 0.0625f)) >> 1, 1);
  const float col0 = (float)(cx - w2), row0 = (float)(cy - h2);
  const float rw   = (float)(2 * w2),  rh   = (float)(2 * h2);
  const float inv14 = 1.0f / (float)RESZ;

  // ---- footprint (samples are monotone; y1 = min(y0+1, 63)) ----
  float ysA = fminf(fmaxf(row0 + 0.5f  * inv14 * rh - 0.5f, 0.f), 63.f);
  float ysB = fminf(fmaxf(row0 + 13.5f * inv14 * rh - 0.5f, 0.f), 63.f);
  float xsA = fminf(fmaxf(col0 + 0.5f  * inv14 * rw - 0.5f, 0.f), 63.f);
  float xsB = fminf(fmaxf(col0 + 13.5f * inv14 * rw - 0.5f, 0.f), 63.f);
  const int ymin = (int)floorf(ysA);
  const int xmin = (int)floorf(xsA);
  const int ymax = min((int)floorf(ysB) + 1, HH - 1);
  const int xmax = min((int)floorf(xsB) + 1, WW - 1);
  const int Ky = min(ymax - ymin + 1, KMAX);
  const int Kx = min(xmax - xmin + 1, KMAX);
  const int cells = Ky * Kx;

  // ---- stage footprint into LDS ----
  if (CHLAST) {
#if HAVE_TDM
    // One TDM descriptor per ROI: 2D tile of the BHWC tensor -> LDS.
    if (tid < 32) {                        // TDM issues once per wave; wave 0 only
      const unsigned long long gaddr =
          (unsigned long long)(uintptr_t)(src +
              ((((long long)b * HH + ymin) * WW) + xmin) * CC);
      const unsigned lds_base = lds_off_u32(&smem[0]);
      u32x4 g0;
      g0.x = 1u;                                        // count=1 (valid), no gather
      g0.y = lds_base;                                  // lds_addr[31:0]
      g0.z = (unsigned)(gaddr & 0xFFFFFFFFull);         // global_addr[31:0]
      g0.w = (unsigned)((gaddr >> 32) & 0x01FFFFFFull)  // global_addr[56:32]
             | 0x80000000u;                             // type=2 ("image")
      i32x8 g1;
      g1[0] = 0x20000;                 // workgroup_mask=0, data_size=2 (4B)
      g1[1] = (int)((unsigned)(WW * CC) << 16);   // tensor_dim0[15:0]=W*C (=0x4000)
      g1[2] = (int)((unsigned)HH << 16);          // dim0[31:16]=0 | tensor_dim1[15:0]=H
      g1[3] = (int)((unsigned)(Kx * CC) << 16);   // dim1[31:16]=0 | tile_dim0=Kx*C
      g1[4] = Ky;                      // tile_dim1=Ky, tile_dim2=0
      g1[5] = WW * CC;                 // tensor_dim0_stride[31:0] = W*C
      g1[6] = 0;                       // dim0_stride[47:32]=0, dim1_stride[15:0]=0
      g1[7] = 0;                       // tensor_dim1_stride[47:16]=0
      const i32x4 gz4 = {};
      const i32x8 gz8 = {};
      __builtin_amdgcn_tensor_load_to_lds(g0, g1, gz4, gz4, gz8, 0);
      wait_tensorcnt0();
    }
#else
    const int l = tid & 63;                // 64 lanes * 16B cover one 1KB cell
    for (int cell = (tid >> 6); cell < cells; cell += 4) {
      const int ky = cell / Kx, kx = cell - ky * Kx;
      const long long gi = ((((long long)b * HH + (ymin + ky)) * WW) + (xmin + kx)) * CC + 4 * l;
      async_ld_b128(lds_off_u32(&smem[cell * CC + 4 * l]),
                    (unsigned long long)(uintptr_t)(src + gi));
    }
    wait_asynccnt0();
#endif
  } else {
    for (int cell = 0; cell < cells; ++cell) {   // lane = channel gather
      const int ky = cell / Kx, kx = cell - ky * Kx;
      const long long gi = ((((long long)b * CC + tid) * HH + (ymin + ky)) * WW) + (xmin + kx);
      async_ld_b32(lds_off_u32(&smem[cell * CC + tid]),
                   (unsigned long long)(uintptr_t)(src + gi));
    }
    wait_asynccnt0();
  }
  __syncthreads();

  // ---- bilinear sample 14x14 grid + 2x2 maxpool, thread = channel ----
  const int c = tid;
  float* outp = out + (size_t)br * (POOLN * POOLN) * CC + c;

  for (int pi = 0; pi < POOLN; ++pi) {
    int   iy0[2], iy1[2];
    float wy[2];
#pragma unroll
    for (int dy = 0; dy < 2; ++dy) {
      const int si = 2 * pi + dy;
      const float ys = fminf(fmaxf(row0 + ((float)si + 0.5f) * inv14 * rh - 0.5f, 0.f), 63.f);
      const int y0 = (int)floorf(ys);
      iy0[dy] = y0 - ymin;
      iy1[dy] = min(y0 + 1, HH - 1) - ymin;
      wy[dy]  = ys - (float)y0;
    }
    for (int pj = 0; pj < POOLN; ++pj) {
      float m = -3.402823466e38f;
#pragma unroll
      for (int dx = 0; dx < 2; ++dx) {
        const int sj = 2 * pj + dx;
        const float xs = fminf(fmaxf(col0 + ((float)sj + 0.5f) * inv14 * rw - 0.5f, 0.f), 63.f);
        const int x0 = (int)floorf(xs);
        const int ix0 = x0 - xmin;
        const int ix1 = min(x0 + 1, WW - 1) - xmin;
        const float wx = xs - (float)x0;
#pragma unroll
        for (int dy = 0; dy < 2; ++dy) {
          const float* rA = smem + (iy0[dy] * Kx) * CC + c;
          const float* rB = smem + (iy1[dy] * Kx) * CC + c;
          const float v00 = rA[ix0 * CC], v01 = rA[ix1 * CC];
          const float v10 = rB[ix0 * CC], v11 = rB[ix1 * CC];
          const float v = (1.0f - wy[dy]) * ((1.0f - wx) * v00 + wx * v01)
                        +         wy[dy]  * ((1.0f - wx) * v10 + wx * v11);
          m = fmaxf(m, v);
        }
      }
      outp[(pi * POOLN + pj) * CC] = m;
    }
  }
}

extern "C" void kernel_launch(void* const* d_in, const int* in_sizes, int n_in,
                              void* d_out, int out_size, void* d_ws, size_t ws_size,
                              hipStream_t stream) {
  const float* fm   = (const float*)d_in[0];   // [B,C,H,W] fp32
  const float* rois = (const float*)d_in[1];   // [B,R,4]   fp32
  float* out = (float*)d_out;                  // [B,R,7,7,C] fp32

  const size_t fm_bytes = (size_t)BB * CC * HH * WW * sizeof(float);

  if (ws_size >= fm_bytes) {
    // channels-last pre-pass -> contiguous 1KB-per-point rows, TDM-friendly
    float* fmT = (float*)d_ws;
    transpose_bchw_to_bhwc<<<dim3(HWN / 16, CC / 16, BB), dim3(16, 16), 0, stream>>>(fm, fmT);
    (void)hipFuncSetAttribute((const void*)roi_align_kernel<true>,
                              hipFuncAttributeMaxDynamicSharedMemorySize, LDS_BYTES);
    roi_align_kernel<true><<<dim3(BB * RR), dim3(256), LDS_BYTES, stream>>>(fmT, rois, out);
  } else {
    (void)hipFuncSetAttribute((const void*)roi_align_kernel<false>,
                              hipFuncAttributeMaxDynamicSharedMemorySize, LDS_BYTES);
    roi_align_kernel<false><<<dim3(BB * RR), dim3(256), LDS_BYTES, stream>>>(fm, rois, out);
  }
}